// NGCF_16355235463443
// MI455X (gfx1250) — compile-verified
//
#include <hip/hip_runtime.h>
#include <hip/hip_bf16.h>

#define N_USERS 100000
#define N_ITEMS 50000
#define N_NODES 150000   // N_USERS + N_ITEMS
#define EMBED   100
#define BATCH   4096

typedef __attribute__((ext_vector_type(2))) float v2f;
typedef __attribute__((ext_vector_type(8))) float v8f;

// ---------------------------------------------------------------- utilities
__global__ void zero_f32(float* __restrict__ p, long n) {
  long i = (long)blockIdx.x * blockDim.x + threadIdx.x;
  if (i < n) p[i] = 0.f;
}

__global__ void concat_feat(const float* __restrict__ u,
                            const float* __restrict__ it,
                            float* __restrict__ feat) {
  const long n1 = (long)N_USERS * EMBED;
  const long n  = (long)N_NODES * EMBED;
  long i = (long)blockIdx.x * blockDim.x + threadIdx.x;
  if (i < n) feat[i] = (i < n1) ? u[i] : it[i - n1];
}

// -------------------------------------------------- SpMM: Y += vals * X[cols]
// One wave32 per edge; lanes stripe the feature dimension. The 60MB feature
// matrix lives in the 192MB L2, so the random gather of X[col] is L2-resident.
// Scatter uses global_atomic_add_f32. Next edge's source row is prefetched
// (global_prefetch_b8).
__global__ void __launch_bounds__(256)
spmm_atomic(const int* __restrict__ rows, const int* __restrict__ cols,
            const float* __restrict__ vals, const float* __restrict__ X,
            float* __restrict__ Y, int F, int nE) {
  int e    = blockIdx.x * (blockDim.x >> 5) + (threadIdx.x >> 5);
  int lane = threadIdx.x & 31;
  if (e >= nE) return;
  int   r = rows[e];
  int   c = cols[e];
  float v = vals[e];
  const float* src = X + (long)c * F;
  float*       dst = Y + (long)r * F;
  if (e + 1 < nE) {
    int cn = cols[e + 1];
    __builtin_prefetch(X + (long)cn * F + lane, 0, 1);
  }
  for (int f = lane; f < F; f += 32)
    atomicAdd(dst + f, v * src[f]);
}

// ------------------------------------------- inter = Lx * X ; sum = Lx + X
__global__ void ew_inter_sum(const float* __restrict__ Lx,
                             const float* __restrict__ X,
                             float* __restrict__ inter,
                             float* __restrict__ sum, long n) {
  long i = (long)blockIdx.x * blockDim.x + threadIdx.x;
  if (i < n) {
    float a = Lx[i], b = X[i];
    inter[i] = a * b;
    sum[i]   = a + b;
  }
}

// ---------------------------------------------------------------- dual GEMM
// OUT = relu(A1 @ W1 + A2 @ W2 + b1 + b2), fp32 throughout via
// V_WMMA_F32_16X16X4_F32. One wave per 16x16 output tile; 8 waves per block.
// W1/W2 are staged in LDS *zero-padded* to Npad = ceil(N/16)*16 columns, so
// the inner loop has NO divergence and no EXEC save/restore between WMMAs.
// A-frag layout (32b 16x4): lanes 0-15 hold K={k0,k0+1}, lanes 16-31 K={k0+2,k0+3}.
__global__ void __launch_bounds__(256)
dual_gemm_bias_relu(const float* __restrict__ A1, const float* __restrict__ A2,
                    const float* __restrict__ W1, const float* __restrict__ W2,
                    const float* __restrict__ b1, const float* __restrict__ b2,
                    float* __restrict__ OUT, int M, int K, int N) {
  __shared__ float sW1[8000];                 // K * Npad <= 100*80
  __shared__ float sW2[8000];
  const int Npad = (N + 15) & ~15;
  const int knp  = K * Npad;
  for (int i = threadIdx.x; i < knp; i += blockDim.x) {
    int k = i / Npad, c = i - k * Npad;
    float w1v = 0.f, w2v = 0.f;
    if (c < N) { w1v = W1[k * N + c]; w2v = W2[k * N + c]; }
    sW1[i] = w1v;
    sW2[i] = w2v;
  }
  __syncthreads();

  int lane    = threadIdx.x & 31;
  int rowTile = blockIdx.x * 8 + (threadIdx.x >> 5);
  if (rowTile * 16 >= M) return;              // uniform per wave -> EXEC all-1s
  int colTile = blockIdx.y;

  int  col   = colTile * 16 + (lane & 15);    // always < Npad
  long m     = (long)rowTile * 16 + (lane & 15);
  int  kOff  = (lane < 16) ? 0 : 2;
  bool colOK = (col < N);

  float bias = colOK ? (b1[col] + b2[col]) : 0.f;
  v8f acc;
#pragma unroll
  for (int r = 0; r < 8; ++r) acc[r] = bias;

  const float* a1p = A1 + m * K;
  const float* a2p = A2 + m * K;
  const float* w1p = sW1 + col;
  const float* w2p = sW2 + col;

  for (int k0 = 0; k0 < K; k0 += 4) {
    int k = k0 + kOff;
    v2f a1; a1.x = a1p[k];           a1.y = a1p[k + 1];
    v2f a2; a2.x = a2p[k];           a2.y = a2p[k + 1];
    v2f wa; wa.x = w1p[k * Npad];    wa.y = w1p[(k + 1) * Npad];
    v2f wb; wb.x = w2p[k * Npad];    wb.y = w2p[(k + 1) * Npad];
    acc = __builtin_amdgcn_wmma_f32_16x16x4_f32(false, a1, false, wa,
                                                (short)0, acc, false, false);
    acc = __builtin_amdgcn_wmma_f32_16x16x4_f32(false, a2, false, wb,
                                                (short)0, acc, false, false);
  }

  if (colOK) {
    long rbase = (long)rowTile * 16 + ((lane < 16) ? 0 : 8);
#pragma unroll
    for (int r = 0; r < 8; ++r) {
      float v = acc[r];
      OUT[(rbase + r) * N + col] = v > 0.f ? v : 0.f;
    }
  }
}

// ---------------------------------------------------- generic GEMM (MLP path)
// OUT = act(A @ W + bias). Out-of-range columns use a clamped address plus a
// select (v_cndmask) instead of a divergent branch, keeping the WMMA loop
// free of EXEC manipulation. Weights are tiny and fully L2-cached.
__global__ void __launch_bounds__(256)
gemm_bias_act(const float* __restrict__ A, const float* __restrict__ W,
              const float* __restrict__ bias, float* __restrict__ OUT,
              int M, int K, int N, int doRelu) {
  int lane    = threadIdx.x & 31;
  int rowTile = blockIdx.x * 8 + (threadIdx.x >> 5);
  if (rowTile * 16 >= M) return;
  int colTile = blockIdx.y;

  int  col   = colTile * 16 + (lane & 15);
  long m     = (long)rowTile * 16 + (lane & 15);
  int  kOff  = (lane < 16) ? 0 : 2;
  bool colOK = (col < N);
  int  colL  = colOK ? col : (N - 1);         // clamped, always in range
  float msk  = colOK ? 1.f : 0.f;

  float b = colOK ? bias[col] : 0.f;
  v8f acc;
#pragma unroll
  for (int r = 0; r < 8; ++r) acc[r] = b;

  const float* ap = A + m * K;
  const float* wp = W + colL;
  for (int k0 = 0; k0 < K; k0 += 4) {
    int k = k0 + kOff;
    v2f a; a.x = ap[k];                  a.y = ap[k + 1];
    v2f w; w.x = wp[(long)k * N] * msk;  w.y = wp[(long)(k + 1) * N] * msk;
    acc = __builtin_amdgcn_wmma_f32_16x16x4_f32(false, a, false, w,
                                                (short)0, acc, false, false);
  }

  if (colOK) {
    long rbase = (long)rowTile * 16 + ((lane < 16) ? 0 : 8);
#pragma unroll
    for (int r = 0; r < 8; ++r) {
      float v = acc[r];
      if (doRelu) v = v > 0.f ? v : 0.f;
      OUT[(rbase + r) * N + col] = v;
    }
  }
}

// ------------------------------------------------- gather [B,460] embeddings
// embd[b] = [feat[u](100) | f1[u](80) | f2[u](50) | feat[i](100) | f1[i](80) | f2[i](50)]
__global__ void __launch_bounds__(256)
gather_embd(const int* __restrict__ userIdx, const int* __restrict__ itemIdx,
            const float* __restrict__ feat, const float* __restrict__ f1,
            const float* __restrict__ f2, float* __restrict__ embd) {
  int b    = blockIdx.x * (blockDim.x >> 5) + (threadIdx.x >> 5);
  int lane = threadIdx.x & 31;
  if (b >= BATCH) return;
  int u = userIdx[b];
  int t = itemIdx[b];            // already offset into node space
  float* dst = embd + (long)b * 460;
  for (int c = lane; c < 460; c += 32) {
    int cc = c, node = u;
    if (cc >= 230) { cc -= 230; node = t; }
    float v;
    if (cc < 100)      v = feat[(long)node * 100 + cc];
    else if (cc < 180) v = f1[(long)node * 80 + (cc - 100)];
    else               v = f2[(long)node * 50 + (cc - 180)];
    dst[c] = v;
  }
}

// ---------------------------------------------------------------- launcher
extern "C" void kernel_launch(void* const* d_in, const int* in_sizes, int n_in,
                              void* d_out, int out_size, void* d_ws, size_t ws_size,
                              hipStream_t stream) {
  const int*   userIdx = (const int*)  d_in[0];
  const int*   itemIdx = (const int*)  d_in[1];
  const int*   rows    = (const int*)  d_in[2];
  const int*   cols    = (const int*)  d_in[3];
  const float* vals    = (const float*)d_in[4];
  const float* uEmbd   = (const float*)d_in[5];
  const float* iEmbd   = (const float*)d_in[6];
  const float* w1  = (const float*)d_in[7];
  const float* b1  = (const float*)d_in[8];
  const float* wi1 = (const float*)d_in[9];
  const float* bi1 = (const float*)d_in[10];
  const float* w2  = (const float*)d_in[11];
  const float* b2  = (const float*)d_in[12];
  const float* wi2 = (const float*)d_in[13];
  const float* bi2 = (const float*)d_in[14];
  const float* t1w = (const float*)d_in[15];
  const float* t1b = (const float*)d_in[16];
  const float* t2w = (const float*)d_in[17];
  const float* t2b = (const float*)d_in[18];
  const float* t3w = (const float*)d_in[19];
  const float* t3b = (const float*)d_in[20];
  float* out = (float*)d_out;
  const int nE = in_sizes[2];    // N_EDGES

  // ---- workspace layout (floats); buffers A/B/C are reused across stages.
  // Total ~ 82M floats (~330 MB).
  float* ws   = (float*)d_ws;
  float* feat = ws;                        // 15,000,000
  float* bufA = feat + 15000000;           // 15,000,000
  float* bufB = bufA + 15000000;           // 15,000,000
  float* bufC = bufB + 15000000;           // 15,000,000
  float* f1   = bufC + 15000000;           // 12,000,000
  float* f2   = f1   + 12000000;           //  7,500,000
  float* embd = f2   + 7500000;            //  1,884,160
  float* h1   = embd + (long)BATCH * 460;  //    262,144
  float* h2   = h1   + (long)BATCH * 64;   //    131,072
  (void)ws_size; (void)n_in; (void)out_size;

  const int TB = 256;
  auto blocks1d = [](long n, int tb) { return (unsigned)((n + tb - 1) / tb); };

  const long nFeat1 = (long)N_NODES * 100;   // 15,000,000
  const long nFeat2 = (long)N_NODES * 80;    // 12,000,000
  const unsigned edgeBlocks = blocks1d(nE, 8);
  const unsigned rowBlkN    = blocks1d(N_NODES / 16, 8);   // 9375 tiles -> 1172
  const unsigned rowBlkB    = blocks1d(BATCH  / 16, 8);    //  256 tiles ->   32

  // features = concat(uEmbd, iEmbd)
  concat_feat<<<blocks1d(nFeat1, TB), TB, 0, stream>>>(uEmbd, iEmbd, feat);

  // ----------------- layer 1 (100 -> 80) -----------------
  zero_f32<<<blocks1d(nFeat1, TB), TB, 0, stream>>>(bufA, nFeat1);
  spmm_atomic<<<edgeBlocks, TB, 0, stream>>>(rows, cols, vals, feat, bufA, 100, nE); // Lx
  ew_inter_sum<<<blocks1d(nFeat1, TB), TB, 0, stream>>>(bufA, feat, bufB, bufC, nFeat1);
  zero_f32<<<blocks1d(nFeat1, TB), TB, 0, stream>>>(bufA, nFeat1);                   // Lx dead
  spmm_atomic<<<edgeBlocks, TB, 0, stream>>>(rows, cols, vals, bufB, bufA, 100, nE); // S2
  dual_gemm_bias_relu<<<dim3(rowBlkN, 5), TB, 0, stream>>>(
      bufC /*Lx+X*/, bufA /*S2*/, w1, wi1, b1, bi1, f1, N_NODES, 100, 80);

  // ----------------- layer 2 (80 -> 50) ------------------
  zero_f32<<<blocks1d(nFeat2, TB), TB, 0, stream>>>(bufB, nFeat2);
  spmm_atomic<<<edgeBlocks, TB, 0, stream>>>(rows, cols, vals, f1, bufB, 80, nE);    // Lx2
  ew_inter_sum<<<blocks1d(nFeat2, TB), TB, 0, stream>>>(bufB, f1, bufC, bufA, nFeat2);
  zero_f32<<<blocks1d(nFeat2, TB), TB, 0, stream>>>(bufB, nFeat2);                   // Lx2 dead
  spmm_atomic<<<edgeBlocks, TB, 0, stream>>>(rows, cols, vals, bufC, bufB, 80, nE);  // S2'
  dual_gemm_bias_relu<<<dim3(rowBlkN, 4), TB, 0, stream>>>(
      bufA /*Lx2+f1*/, bufB /*S2'*/, w2, wi2, b2, bi2, f2, N_NODES, 80, 50);

  // ----------------- gather + MLP ------------------------
  gather_embd<<<blocks1d(BATCH, 8), TB, 0, stream>>>(userIdx, itemIdx, feat, f1, f2, embd);
  gemm_bias_act<<<dim3(rowBlkB, 4), TB, 0, stream>>>(embd, t1w, t1b, h1, BATCH, 460, 64, 1);
  gemm_bias_act<<<dim3(rowBlkB, 2), TB, 0, stream>>>(h1,   t2w, t2b, h2, BATCH, 64,  32, 0);
  gemm_bias_act<<<dim3(rowBlkB, 1), TB, 0, stream>>>(h2,   t3w, t3b, out, BATCH, 32,  1, 0);
}